// TimingEngine_1211180778106
// MI455X (gfx1250) — compile-verified
//
#include <hip/hip_runtime.h>
#include <math.h>

// ---------------------------------------------------------------------------
// Static timing engine for MI455X (gfx1250, wave32).
// Gather/latency-bound workload: per-arc 8x8 LUT bilinear + ceff iteration.
// LUTs (10.2MB) + axes (2.5MB) are L2-resident (192MB L2). No matmul
// structure exists, so WMMA is inapplicable; we lean on wave32 occupancy,
// WGP-scope global_prefetch_b8 (inline asm -- the builtin can only reach
// SE/SYS scope, which skip the WGP cache), and minimal-footprint gathers.
// ---------------------------------------------------------------------------

#define SAXIS 8
#define CAXIS 8
#define SCALE_F 1.0e15f
#define MAX_CAP_F 1.0e-12f
#define MAX_ITER 3

// WGP-scope prefetch: scope field omitted -> SCOPE=0 (WGP), which pulls the
// line into all cache levels (ISA 10.5). clang's __builtin_prefetch can only
// emit SCOPE_SE/SCOPE_SYS, which do no WGP-level prefetch.
__device__ __forceinline__ void prefetch_wgp(const void* p) {
    asm volatile("global_prefetch_b8 %0, off" :: "v"((unsigned long long)p));
}

// i0 = clip(count(ax <= x) - 1, 0, 6); also return ax[i0], ax[i0+1]
// via an unrolled cndmask chain (no scratch, no dynamic VGPR indexing).
__device__ __forceinline__ int interval8(const float ax[8], float x,
                                         float& x0, float& x1) {
    int cnt = 0;
#pragma unroll
    for (int k = 0; k < 8; ++k) cnt += (ax[k] <= x) ? 1 : 0;
    int i0 = cnt - 1;
    i0 = i0 < 0 ? 0 : (i0 > 6 ? 6 : i0);
    x0 = ax[0]; x1 = ax[1];
#pragma unroll
    for (int k = 1; k < 7; ++k) {
        bool p = (i0 == k);
        x0 = p ? ax[k] : x0;
        x1 = p ? ax[k + 1] : x1;
    }
    return i0;
}

__device__ __forceinline__ void load8(const float* __restrict__ p, float v[8]) {
    float4 a = ((const float4*)p)[0];   // global_load_b128
    float4 b = ((const float4*)p)[1];
    v[0] = a.x; v[1] = a.y; v[2] = a.z; v[3] = a.w;
    v[4] = b.x; v[5] = b.y; v[6] = b.z; v[7] = b.w;
}

// One sub-arc: ceff fixed point (3 iters) + final bilinear delay/slew.
__device__ __forceinline__ void sub_arc(
    int arc, float xs, float xa, float c1f, float c2f,
    const float* __restrict__ dtab, const float* __restrict__ stab,
    const float* __restrict__ sidx, const float* __restrict__ lidx,
    float& arr_out, float& slew_out)
{
    float axs[8], axc[8];
    load8(sidx + (size_t)arc * SAXIS, axs);
    load8(lidx + (size_t)arc * CAXIS, axc);

    // slew axis interval (fixed across iterations)
    float sx0, sx1;
    int i0 = interval8(axs, xs, sx0, sx1);
    float a = (xs - sx0) / (sx1 - sx0);
    float one_m_a = 1.0f - a;

    const float* db = dtab + (size_t)arc * (SAXIS * CAXIS) + i0 * CAXIS;
    const float* sb = stab + (size_t)arc * (SAXIS * CAXIS) + i0 * CAXIS;

    float denom_s = fmaxf(xs, 1e-30f);
    float ceff = fmaxf(c1f + c2f, 1e-30f);

#pragma unroll
    for (int it = 0; it < MAX_ITER; ++it) {
        float cx0, cx1;
        int j0 = interval8(axc, ceff, cx0, cx1);
        float b = (ceff - cx0) / (cx1 - cx0);
        float t00 = db[j0],         t01 = db[j0 + 1];
        float t10 = db[CAXIS + j0], t11 = db[CAXIS + j0 + 1];
        float d = one_m_a * (1.0f - b) * t00 + one_m_a * b * t01
                + a * (1.0f - b) * t10 + a * b * t11;
        float tau   = fmaxf(d, 1e-30f);
        float ratio = fminf(2.0f * tau / denom_s, 10.0f);
        float h = (ratio > 0.01f) ? (1.0f - expf(-ratio)) / ratio
                                  : 1.0f - 0.5f * ratio;
        ceff = fmaxf(c1f + c2f * h, 1e-30f);
    }

    float load = fminf(ceff, MAX_CAP_F);
    float cx0, cx1;
    int j0 = interval8(axc, load, cx0, cx1);
    float b = (load - cx0) / (cx1 - cx0);
    float w00 = one_m_a * (1.0f - b), w01 = one_m_a * b;
    float w10 = a * (1.0f - b),       w11 = a * b;

    float d = w00 * db[j0] + w01 * db[j0 + 1]
            + w10 * db[CAXIS + j0] + w11 * db[CAXIS + j0 + 1];
    float s = w00 * sb[j0] + w01 * sb[j0 + 1]
            + w10 * sb[CAXIS + j0] + w11 * sb[CAXIS + j0 + 1];

    arr_out  = xa + d;
    slew_out = s;
}

__global__ void __launch_bounds__(256)
timing_kernel(const float* __restrict__ in_arr,
              const float* __restrict__ in_slew,
              const float* __restrict__ c1,
              const float* __restrict__ c2,
              const int* __restrict__ arc_r,
              const int* __restrict__ arc_f,
              const int* __restrict__ unate,
              const float* __restrict__ dtab,
              const float* __restrict__ stab,
              const float* __restrict__ sidx,
              const float* __restrict__ lidx,
              float* __restrict__ out, int T)
{
    int t = blockIdx.x * blockDim.x + threadIdx.x;
    if (t >= T) return;

    int a0 = arc_r[t];
    int a1 = arc_f[t];

    // WGP-scope prefetch: prime L0/WGP$ with the LUT cachelines (already
    // L2-resident) while we do the unateness lookup and axis loads,
    // hiding the L2->L0 latency of the dependent gather chain.
    prefetch_wgp(dtab + (size_t)a0 * 64);
    prefetch_wgp(dtab + (size_t)a1 * 64);
    prefetch_wgp(stab + (size_t)a0 * 64);
    prefetch_wgp(stab + (size_t)a1 * 64);

    // rf select: col 0 uses rise arc (is_fall=0), col 1 uses fall arc (is_fall=1)
    int rf0 = unate[a0] & 1;         // una ^ 0
    int rf1 = (unate[a1] ^ 1) & 1;   // una ^ 1

    float2 sl = ((const float2*)in_slew)[t];
    float2 ar = ((const float2*)in_arr)[t];
    float xs0 = rf0 ? sl.y : sl.x;
    float xa0 = rf0 ? ar.y : ar.x;
    float xs1 = rf1 ? sl.y : sl.x;
    float xa1 = rf1 ? ar.y : ar.x;

    float c1f = c1[t] / SCALE_F;
    float c2f = c2[t] / SCALE_F;

    float arr0, slw0, arr1, slw1;
    sub_arc(a0, xs0, xa0, c1f, c2f, dtab, stab, sidx, lidx, arr0, slw0);
    sub_arc(a1, xs1, xa1, c1f, c2f, dtab, stab, sidx, lidx, arr1, slw1);

    float4 o;
    o.x = arr0; o.y = arr1; o.z = slw0; o.w = slw1;
    ((float4*)out)[t] = o;
}

// Deterministic per-gate softmax exploiting sorted gate_id: each segment head
// serially reduces its (avg length 4) contiguous segment. No atomics, so the
// result is bit-identical on every replay.
__global__ void __launch_bounds__(256)
softmax_kernel(const float* __restrict__ U,
               const int* __restrict__ gid,
               float* __restrict__ w, int n)
{
    int i = blockIdx.x * blockDim.x + threadIdx.x;
    if (i >= n) return;
    int g = gid[i];
    if (i > 0 && gid[i - 1] == g) return;     // not a segment head

    int end = i + 1;
    while (end < n && gid[end] == g) ++end;

    float m = -__builtin_inff();
    for (int j = i; j < end; ++j) m = fmaxf(m, U[j]);
    float s = 0.0f;
    for (int j = i; j < end; ++j) s += expf(U[j] - m);
    for (int j = i; j < end; ++j) w[j] = expf(U[j] - m) / s;
}

extern "C" void kernel_launch(void* const* d_in, const int* in_sizes, int n_in,
                              void* d_out, int out_size, void* d_ws, size_t ws_size,
                              hipStream_t stream)
{
    const float* U       = (const float*)d_in[0];
    const int*   gid     = (const int*)  d_in[1];
    const float* in_arr  = (const float*)d_in[2];
    const float* in_slew = (const float*)d_in[3];
    const float* c1      = (const float*)d_in[4];
    const float* c2      = (const float*)d_in[5];
    // d_in[6] = rpi : dead in the reference computation, skipped
    const int*   arc_r   = (const int*)  d_in[7];
    const int*   arc_f   = (const int*)  d_in[8];
    const int*   unate   = (const int*)  d_in[9];
    const float* dtab    = (const float*)d_in[10];
    const float* stab    = (const float*)d_in[11];
    const float* sidx    = (const float*)d_in[12];   // slew_index [A,8]
    const float* lidx    = (const float*)d_in[13];   // load_index [A,8]

    const int T  = in_sizes[4];   // c1 has length T
    const int Un = in_sizes[0];   // U_flat length

    float* out     = (float*)d_out;          // [T,4] flat
    float* weights = out + (size_t)T * 4;    // [TOTAL_U]

    timing_kernel<<<(T + 255) / 256, 256, 0, stream>>>(
        in_arr, in_slew, c1, c2, arc_r, arc_f, unate,
        dtab, stab, sidx, lidx, out, T);

    softmax_kernel<<<(Un + 255) / 256, 256, 0, stream>>>(U, gid, weights, Un);
}